// Local_Dependency_layer_87548613362053
// MI455X (gfx1250) — compile-verified
//
#include <hip/hip_runtime.h>

typedef float v2f __attribute__((ext_vector_type(2)));
typedef float v8f __attribute__((ext_vector_type(8)));

#define P_DIM     512
#define E_DIM     64
#define NE_STRIDE 32768     /* P_DIM*E_DIM : stride between (b,c) columns      */
#define NT        64        /* workgroup n-tile (wave computes all 64 cols)    */
#define ET        8         /* e values per workgroup (1 per wave)             */
#define KC        16        /* q-chunk staged per LDS buffer                   */
#define NCHUNK    32        /* 512 / 16                                        */
/* staging layout: idx = (q>>1)*SQH + n*SN2 + e*2 + (q&1)                      */
/*  - K-pair contiguous -> B frag is one aligned ds_load_b64                   */
/*  - SN2=18: 18*dn==0 mod 64 => dn==0 mod 32 -> 16 distinct banks per group   */
/*  - SQH==32 mod 64: 18*dn==32 => dn==16, unsolvable -> groups disjoint       */
#define SN2       18
#define SQH       1184      /* 64*18 + 32 */
#define BUFSZ     9472      /* 8 * SQH floats per buffer (~37 KB, x2 = 74 KB)  */
/* epilogue layout: (p*64 + n)*9 + e  (max 9215 < BUFSZ)                       */
#define EPS       9

__global__ __launch_bounds__(256)
void perchan_linear_wmma(const float* __restrict__ X,
                         const float* __restrict__ Wt,
                         const float* __restrict__ Bias,
                         float* __restrict__ Out) {
  __shared__ float lds[2][BUFSZ];

  const int tid  = threadIdx.x;
  const int lane = tid & 31;
  const int wave = tid >> 5;       // 8 waves; wave w owns e = e0 + w
  const int m    = lane & 15;      // M-row for A, N-col (mod 16) for B/C
  const int g    = lane >> 4;      // half-wave group

  const int n0 = blockIdx.x * NT;  // (b,c) tile
  const int p0 = blockIdx.y * 16;  // p tile
  const int e0 = blockIdx.z * ET;  // e group (8 per workgroup)
  const int e  = e0 + wave;

  // ---- 4 accumulators (n-subtiles) seeded with bias b[e, p0 + r + 8g] ----
  v8f acc[4];
  {
    const float* bp = Bias + (size_t)e * P_DIM + p0 + 8 * g;
#pragma unroll
    for (int r = 0; r < 8; ++r) {
      const float bv = bp[r];
      acc[0][r] = bv; acc[1][r] = bv; acc[2][r] = bv; acc[3][r] = bv;
    }
  }

  // ---- per-lane W pointer (A frag): VGPR0 -> K=2g, VGPR1 -> K=2g+1 ----
  const float* wp = Wt + (size_t)e * (P_DIM * P_DIM)
                       + (size_t)(p0 + m) * P_DIM + 2 * g;

  const float* xb = X + (size_t)n0 * NE_STRIDE + e0;

  // ---- cooperative x-tile loader coords: 2048 float4s / 256 threads ----
  int ceq[8], cn[8], cq[8], cst[8];
#pragma unroll
  for (int i = 0; i < 8; ++i) {
    const int v = tid + 256 * i;
    const int eq = v & 1;          // which float4 of the 8-e run
    const int n  = (v >> 1) & 63;  // n within tile
    const int q  = v >> 7;         // q within chunk
    ceq[i] = eq * 4;
    cn[i]  = n;
    cq[i]  = q;
    // LDS base for this item: elements go at base + 2*t (t = 0..3 over e)
    cst[i] = (q >> 1) * SQH + n * SN2 + eq * 8 + (q & 1);
  }

  float4 xv[8];
  // prologue: stage chunk 0
#pragma unroll
  for (int i = 0; i < 8; ++i)
    xv[i] = *(const float4*)(xb + (size_t)cn[i] * NE_STRIDE
                                + (size_t)cq[i] * E_DIM + ceq[i]);
#pragma unroll
  for (int i = 0; i < 8; ++i) {
    float* d = &lds[0][cst[i]];
    d[0] = xv[i].x; d[2] = xv[i].y; d[4] = xv[i].z; d[6] = xv[i].w;
  }
  __syncthreads();

#pragma unroll 1
  for (int c = 0; c < NCHUNK; ++c) {
    const int buf = c & 1;
    // prefetch next chunk into registers (coalesced over e)
    if (c + 1 < NCHUNK) {
#pragma unroll
      for (int i = 0; i < 8; ++i)
        xv[i] = *(const float4*)(xb + (size_t)cn[i] * NE_STRIDE
                                    + (size_t)((c + 1) * KC + cq[i]) * E_DIM + ceq[i]);
    }
    // compute: 4 K-steps x 4 n-subtiles => 16 v_wmma, A frag reused 4x,
    // each B frag is one aligned ds_load_b64 straight into a VGPR pair
    const float* w  = wp + c * KC;
    const float* Bt = lds[buf];
#pragma unroll
    for (int s = 0; s < 4; ++s) {
      const v2f a = *(const v2f*)(w + s * 4);        // global_load_b64
      const int qh = s * 2 + g;                      // (4s + 2g) >> 1
#pragma unroll
      for (int u = 0; u < 4; ++u) {
        const int nn = u * 16 + m;
        const v2f b = *(const v2f*)&Bt[qh * SQH + nn * SN2 + wave * 2];
        acc[u] = __builtin_amdgcn_wmma_f32_16x16x4_f32(
                     false, a, false, b, (short)0, acc[u], false, false);
      }
    }
    // store prefetched chunk to the other buffer (ping-pong: 1 barrier/iter)
    if (c + 1 < NCHUNK) {
      const int nb = buf ^ 1;
#pragma unroll
      for (int i = 0; i < 8; ++i) {
        float* d = &lds[nb][cst[i]];
        d[0] = xv[i].x; d[2] = xv[i].y; d[4] = xv[i].z; d[6] = xv[i].w;
      }
    }
    __syncthreads();
  }

  // ---- epilogue: gather C tiles in LDS, write coalesced float4 over e ----
#pragma unroll
  for (int u = 0; u < 4; ++u) {
#pragma unroll
    for (int r = 0; r < 8; ++r) {
      const int p = r + 8 * g;               // C row (p local)
      lds[0][(p * NT + u * 16 + m) * EPS + wave] = acc[u][r];
    }
  }
  __syncthreads();

  float* ob = Out + (size_t)n0 * NE_STRIDE + (size_t)p0 * E_DIM + e0;
#pragma unroll
  for (int i = 0; i < 8; ++i) {
    const int v  = tid + 256 * i;
    const int eq = (v & 1) * 4;
    const int n  = (v >> 1) & 63;
    const int p  = v >> 7;
    const float* s = &lds[0][(p * NT + n) * EPS + eq];
    float4 o; o.x = s[0]; o.y = s[1]; o.z = s[2]; o.w = s[3];
    *(float4*)(ob + (size_t)n * NE_STRIDE + (size_t)p * E_DIM + eq) = o;
  }
}

extern "C" void kernel_launch(void* const* d_in, const int* in_sizes, int n_in,
                              void* d_out, int out_size, void* d_ws, size_t ws_size,
                              hipStream_t stream) {
  (void)in_sizes; (void)n_in; (void)d_ws; (void)ws_size; (void)out_size;
  const float* x  = (const float*)d_in[0];   // (16,128,512,64)
  const float* W  = (const float*)d_in[1];   // (64,512,512)
  const float* b  = (const float*)d_in[2];   // (64,512)
  float* out      = (float*)d_out;           // (16,128,512,64)

  dim3 grid(2048 / NT, P_DIM / 16, E_DIM / ET);  // (32, 32, 8)
  dim3 block(256);
  hipLaunchKernelGGL(perchan_linear_wmma, grid, block, 0, stream, x, W, b, out);
}